// PixelShuffle_31610959299036
// MI455X (gfx1250) — compile-verified
//
#include <hip/hip_runtime.h>
#include <stdint.h>

// PixelShuffle r=2 on [16,128,128,256] f32 -> [16,256,256,64] f32.
// Pure permutation: 256MB in + 256MB out @ 23.3 TB/s => ~22us roofline.
// Strategy: async global->LDS staged copy (CDNA5 ASYNCcnt path) with NT load
// hints (stream > L2 capacity), bank-conflict-free LDS transpose, and
// non-temporal B128 output stores.

typedef float v4f __attribute__((ext_vector_type(4)));

#define THREADS          256
#define CHUNKS_PER_TILE  8      // 8 x 1KB input chunks per tile
#define TILES_PER_BLOCK  8
#define NUM_BLOCKS       4096   // 4096*8 = 32768 tiles = 262144 chunks total

// LDS geometry (floats): each 64B group of 16 data floats padded to 80B.
// -> async B128 writes stay 16B aligned; ds_load_b64 lane stride = 20 dwords
//    which hits 32 distinct banks across a wave (no conflicts).
#define GRP_STRIDE_F    20
#define CHUNK_STRIDE_F  320     // 16 groups * 20
#define BUF_STRIDE_F    2560    // 8 chunks * 320
#define SMEM_F          (2 * BUF_STRIDE_F)   // 20 KB double-buffered

__device__ __forceinline__ void async_load_b128(unsigned lds_addr, const void* gaddr) {
    // Input is read exactly once (256MB stream > 192MB L2): non-temporal.
    asm volatile("global_load_async_to_lds_b128 %0, %1, off th:TH_LOAD_NT"
                 :: "v"(lds_addr), "v"((unsigned long long)(uintptr_t)gaddr)
                 : "memory");
}
__device__ __forceinline__ void wait_async_le2() {
    asm volatile("s_wait_asynccnt 0x2" ::: "memory");
}
__device__ __forceinline__ void wait_async_0() {
    asm volatile("s_wait_asynccnt 0x0" ::: "memory");
}

__device__ __forceinline__ void issue_tile(const float* __restrict__ in,
                                           unsigned lds_base, int buf, int tile, int tid) {
    // Tile = 8 chunks = 2048 floats = 512 x 16B units. 256 threads x 2 units.
    const char* gbase = (const char*)(in + (size_t)tile * 2048);
#pragma unroll
    for (int s = 0; s < 2; ++s) {
        int u     = tid + THREADS * s;      // 0..511, consecutive per wave -> 512B/instr
        int chunk = u >> 6;                 // 64 units per chunk
        int uu    = u & 63;
        unsigned lds = lds_base
                     + (unsigned)(buf * (BUF_STRIDE_F * 4)
                                  + chunk * (CHUNK_STRIDE_F * 4)
                                  + (uu >> 2) * (GRP_STRIDE_F * 4)
                                  + (uu & 3) * 16);
        async_load_b128(lds, gbase + (size_t)u * 16);
    }
}

__device__ __forceinline__ void process_tile(const float* __restrict__ smem,
                                             float* __restrict__ out,
                                             int buf, int tile, int tid) {
    // Work unit: (chunk-in-tile, ccq, h). One wave handles one chunk.
    int chunkloc = tid >> 5;          // 0..7
    int lane     = tid & 31;
    int ccq      = lane >> 1;         // 0..15 : output channel block of 4
    int h        = lane & 1;          // ra = h

    int m = tile * CHUNKS_PER_TILE + chunkloc;  // global chunk id = ((n*128)+i)*128+j
    int j = m & 127;
    int i = (m >> 7) & 127;
    int n = m >> 14;

    const float* g = smem + buf * BUF_STRIDE_F + chunkloc * CHUNK_STRIDE_F
                          + ccq * GRP_STRIDE_F;
    // element e = 16*ccq + 4*q + k lives at group base + q*4 + k (floats).
    // k index = 2*ra + rb ; this thread takes ra = h, so k in {2h, 2h+1}.
    float2 v0 = *(const float2*)(g + 0 * 4 + 2 * h);
    float2 v1 = *(const float2*)(g + 1 * 4 + 2 * h);
    float2 v2 = *(const float2*)(g + 2 * 4 + 2 * h);
    float2 v3 = *(const float2*)(g + 3 * 4 + 2 * h);

    unsigned oj  = 2u * j + h;        // ra = h
    unsigned cc  = 4u * ccq;
    unsigned row0 = (unsigned)n * 256u + 2u * i;       // rb = 0  (.x lane)
    unsigned idx0 = (row0 * 256u + oj) * 64u + cc;
    unsigned idx1 = ((row0 + 1u) * 256u + oj) * 64u + cc;  // rb = 1 (.y lane)

    v4f w0 = { v0.x, v1.x, v2.x, v3.x };
    v4f w1 = { v0.y, v1.y, v2.y, v3.y };
    // Streaming output (written once, never reused) -> non-temporal B128.
    __builtin_nontemporal_store(w0, (v4f*)(out + idx0));
    __builtin_nontemporal_store(w1, (v4f*)(out + idx1));
}

__global__ __launch_bounds__(THREADS)
void pixel_shuffle_async_kernel(const float* __restrict__ in, float* __restrict__ out) {
    __shared__ float smem[SMEM_F];
    const int tid = threadIdx.x;
    // Flat->LDS: low 32 bits of the generic pointer are the DS byte address.
    const unsigned lds_base = (unsigned)(uintptr_t)&smem[0];

    const int tile0 = blockIdx.x * TILES_PER_BLOCK;

    issue_tile(in, lds_base, 0, tile0, tid);
#pragma unroll 1
    for (int t = 0; t < TILES_PER_BLOCK; ++t) {
        const int buf = t & 1;
        if (t + 1 < TILES_PER_BLOCK) {
            issue_tile(in, lds_base, buf ^ 1, tile0 + t + 1, tid);
            wait_async_le2();   // loads retire in order: tile t is resident
        } else {
            wait_async_0();
        }
        __syncthreads();        // all waves' async writes visible
        process_tile(smem, out, buf, tile0 + t, tid);
        __syncthreads();        // release buffer before it is overwritten
    }
}

extern "C" void kernel_launch(void* const* d_in, const int* in_sizes, int n_in,
                              void* d_out, int out_size, void* d_ws, size_t ws_size,
                              hipStream_t stream) {
    const float* in = (const float*)d_in[0];   // [16,128,128,256] f32
    // d_in[1] is r == 2 (hardcoded in the kernel's index math)
    float* out = (float*)d_out;                // [16,256,256,64] f32
    pixel_shuffle_async_kernel<<<NUM_BLOCKS, THREADS, 0, stream>>>(in, out);
}